// YOLO_32564442038555
// MI455X (gfx1250) — compile-verified
//
#include <hip/hip_runtime.h>
#include <hip/hip_bf16.h>
#include <math.h>

// ---------------------------------------------------------------------------
// YOLO post-processing for MI455X (gfx1250, wave32).
//   1) score_kernel : per-anchor max-logit -> sigmoid score + argmax label
//   2) radix-select : exact 1000th-largest 64-bit key per level (8x 256-bin)
//   3) compact+sort : gather the exactly-1000 keys, bitonic-sort 1024 in LDS
//   4) decode       : class-offset boxes, NMS keys, bbox/score/label outputs
//   5) gsort/gather : global descending sort of 3000 candidates (4096 bitonic)
//   6) mask_k       : TDM bulk-stages the 48KB sorted-box table into LDS
//                     (tensor_load_to_lds + s_wait_tensorcnt); 16x16 IoU tiles
//                     with area_i+area_j via V_WMMA_F32_16X16X4_F32; bits via
//                     ballot_w32
//   7) reduce_k     : exact greedy-NMS reduction, single-wave workgroup
// ---------------------------------------------------------------------------

typedef unsigned int       u32;
typedef unsigned long long u64;
typedef __attribute__((ext_vector_type(2))) float v2f;
typedef __attribute__((ext_vector_type(8))) float v8f;
typedef __attribute__((ext_vector_type(4))) unsigned int u32x4;
typedef __attribute__((ext_vector_type(8))) int          i32x8;
typedef __attribute__((ext_vector_type(4))) int          i32x4;

#define N_ANCH   85936      // 65536 + 16384 + 4096
#define N_CAND   3000
#define N_PAD    3008       // 188 * 16
#define NTILE    188
#define WORDS    94         // ceil(3008/32)
#define CONF_T   0.05f
#define NMS_T    0.6f
#define CLS_OFF  8192.0f    // 4.0 * IMG_W

__device__ __forceinline__ int lvl_M(int l)   { return l == 0 ? 65536 : (l == 1 ? 16384 : 4096); }
__device__ __forceinline__ int lvl_off(int l) { return l == 0 ? 0     : (l == 1 ? 65536 : 81920); }
__device__ __forceinline__ int lvl_W(int l)   { return l == 0 ? 256   : (l == 1 ? 128   : 64); }
__device__ __forceinline__ float lvl_S(int l) { return l == 0 ? 8.f   : (l == 1 ? 16.f  : 32.f); }

__device__ __forceinline__ u32 f2ord(float f) {
    u32 u = __float_as_uint(f);
    return (u & 0x80000000u) ? ~u : (u | 0x80000000u);
}
__device__ __forceinline__ float ord2f(u32 u) {
    u32 b = (u & 0x80000000u) ? (u ^ 0x80000000u) : ~u;
    return __uint_as_float(b);
}
__device__ __forceinline__ float sigmoidf(float x) { return 1.f / (1.f + expf(-x)); }

// ---------------------------------------------------------------------------
// Workspace layout (bytes, 16-aligned)
// ---------------------------------------------------------------------------
#define OFF_SCORES  0u          // 85936 f32
#define OFF_LABELS  343744u     // 85936 i32
#define OFF_HIST    687488u     // 8 passes * 3 levels * 256 u32
#define OFF_STATE   712128u     // prefix u64[3] | k u32[3] | cnt u32[3]
#define OFF_SEL     712192u     // 3 * 1024 u64
#define OFF_OBOX    736768u     // 3000*4 f32 (class-offset boxes)
#define OFF_VALID   784768u     // 3000 u32
#define OFF_GKEYS   796768u     // 4096 u64
#define OFF_ORDER   829536u     // 3008 u32 (rank -> slot)
#define OFF_SBOX    841568u     // 3008*4 f32 (sorted offset boxes)
#define OFF_SVALID  889696u     // 3008 u32
#define OFF_MASK    901728u     // 3008 * 94 u32 suppression bitmask
// total ~2.03 MB

// ---------------------------------------------------------------------------
__global__ void init_ws(u32* hist, u32* cnt, u64* prefix, u32* krem,
                        u64* sel, u64* gkeys, u32* mask) {
    int i = blockIdx.x * blockDim.x + threadIdx.x;
    int stride = gridDim.x * blockDim.x;
    for (int t = i; t < N_PAD * WORDS; t += stride) mask[t] = 0u;
    for (int t = i; t < 8 * 3 * 256;   t += stride) hist[t] = 0u;
    for (int t = i; t < 3 * 1024;      t += stride) sel[t]  = 0ull;
    for (int t = i; t < 4096;          t += stride) gkeys[t] = 0ull;
    if (i < 3) { cnt[i] = 0u; krem[i] = 1000u; prefix[i] = 0ull; }
}

// ---------------------------------------------------------------------------
// 1) scores + labels.  max over sigmoid == sigmoid(max), argmax identical.
// ---------------------------------------------------------------------------
__global__ __launch_bounds__(256)
void score_kernel(const float* __restrict__ cls0, const float* __restrict__ cls1,
                  const float* __restrict__ cls2,
                  float* __restrict__ scores, int* __restrict__ labels) {
    int a = blockIdx.x * 256 + threadIdx.x;
    if (a >= N_ANCH) return;
    const float* cls; int m;
    if (a < 65536)      { cls = cls0; m = a; }
    else if (a < 81920) { cls = cls1; m = a - 65536; }
    else                { cls = cls2; m = a - 81920; }
    const float4* row = (const float4*)(cls + (size_t)m * 80);
    __builtin_prefetch(row + 20, 0, 1);   // next row -> global_prefetch_b8
    float best = -3.4e38f; int bi = 0;
#pragma unroll
    for (int q = 0; q < 20; ++q) {
        float4 v = row[q];
        int c = q * 4;
        if (v.x > best) { best = v.x; bi = c;     }
        if (v.y > best) { best = v.y; bi = c + 1; }
        if (v.z > best) { best = v.z; bi = c + 2; }
        if (v.w > best) { best = v.w; bi = c + 3; }
    }
    scores[a] = sigmoidf(best);
    labels[a] = bi;
}

// ---------------------------------------------------------------------------
// 2) 64-bit radix select: key = (ordered_score<<32) | ~index (unique keys,
//    lax.top_k stable tie order).  8 passes x 256 bins, 3 levels via grid.y.
// ---------------------------------------------------------------------------
__global__ __launch_bounds__(256)
void radix_hist(const float* __restrict__ scores, const u64* __restrict__ prefix,
                u32* __restrict__ hist, int pass) {
    __shared__ u32 lh[256];
    int lvl = blockIdx.y;
    int M = lvl_M(lvl), off = lvl_off(lvl);
    lh[threadIdx.x] = 0u;
    __syncthreads();
    int idx = blockIdx.x * 256 + threadIdx.x;
    if (idx < M) {
        u64 key = ((u64)f2ord(scores[off + idx]) << 32) | (u64)(0xFFFFFFFFu - (u32)idx);
        bool match = (pass == 0) ||
                     ((key >> (64 - 8 * pass)) == (prefix[lvl] >> (64 - 8 * pass)));
        if (match) atomicAdd(&lh[(u32)(key >> (56 - 8 * pass)) & 255u], 1u);
    }
    __syncthreads();
    if (lh[threadIdx.x]) atomicAdd(&hist[(pass * 3 + lvl) * 256 + threadIdx.x], lh[threadIdx.x]);
}

__global__ void radix_scan(const u32* __restrict__ hist, u64* __restrict__ prefix,
                           u32* __restrict__ krem, int pass) {
    int lvl = threadIdx.x;
    if (lvl >= 3) return;
    u32 k = krem[lvl];
    const u32* h = hist + (pass * 3 + lvl) * 256;
    u32 cum = 0; int b = 0;
    for (b = 255; b > 0; --b) {
        u32 c = h[b];
        if (cum + c >= k) break;
        cum += c;
    }
    krem[lvl] = k - cum;
    prefix[lvl] |= ((u64)(u32)b) << (56 - 8 * pass);
}

// prefix == exact 1000th-largest key T; unique keys => #(>=T) == 1000
__global__ __launch_bounds__(256)
void compact_k(const float* __restrict__ scores, const u64* __restrict__ prefix,
               u32* __restrict__ cnt, u64* __restrict__ sel) {
    int lvl = blockIdx.y;
    int M = lvl_M(lvl), off = lvl_off(lvl);
    int idx = blockIdx.x * 256 + threadIdx.x;
    if (idx >= M) return;
    u64 key = ((u64)f2ord(scores[off + idx]) << 32) | (u64)(0xFFFFFFFFu - (u32)idx);
    if (key >= prefix[lvl]) {
        u32 p = atomicAdd(&cnt[lvl], 1u);
        if (p < 1024u) sel[lvl * 1024 + p] = key;
    }
}

// ---------------------------------------------------------------------------
// 3) LDS bitonic sorts (descending); pad keys (0) sink to the back.
// ---------------------------------------------------------------------------
__device__ __forceinline__ void bitonic_desc(u64* s, int n, int tid, int nth) {
    for (int k = 2; k <= n; k <<= 1) {
        for (int j = k >> 1; j > 0; j >>= 1) {
            __syncthreads();
            for (int i = tid; i < n; i += nth) {
                int p = i ^ j;
                if (p > i) {
                    bool up = ((i & k) == 0);
                    u64 a = s[i], b = s[p];
                    if (up ? (a < b) : (a > b)) { s[i] = b; s[p] = a; }
                }
            }
        }
    }
    __syncthreads();
}

__global__ __launch_bounds__(512)
void sort1024(u64* __restrict__ sel) {
    __shared__ u64 s[1024];
    int lvl = blockIdx.x, tid = threadIdx.x;
    for (int i = tid; i < 1024; i += 512) s[i] = sel[lvl * 1024 + i];
    bitonic_desc(s, 1024, tid, 512);
    for (int i = tid; i < 1024; i += 512) sel[lvl * 1024 + i] = s[i];
}

__global__ __launch_bounds__(1024)
void gsort(u64* __restrict__ gkeys) {
    __shared__ u64 s[4096];
    int tid = threadIdx.x;
    for (int i = tid; i < 4096; i += 1024) s[i] = gkeys[i];
    bitonic_desc(s, 4096, tid, 1024);
    for (int i = tid; i < 4096; i += 1024) gkeys[i] = s[i];
}

// ---------------------------------------------------------------------------
// 4) decode selected candidates; write bbox/score/label outputs directly.
// ---------------------------------------------------------------------------
__global__ __launch_bounds__(256)
void decode_k(const u64* __restrict__ sel,
              const float* __restrict__ reg0, const float* __restrict__ reg1,
              const float* __restrict__ reg2, const float* __restrict__ scales,
              const int* __restrict__ labels_all,
              float* __restrict__ obox, u32* __restrict__ valid,
              u64* __restrict__ gkeys, float* __restrict__ out) {
    int s = blockIdx.x * 256 + threadIdx.x;
    if (s >= N_CAND) return;
    int lvl = s / 1000, r = s % 1000;
    u64 key = sel[lvl * 1024 + r];
    u32 idx = 0xFFFFFFFFu - (u32)(key & 0xFFFFFFFFull);
    u32 sb  = (u32)(key >> 32);
    float score = ord2f(sb);
    int   W = lvl_W(lvl);
    float st = lvl_S(lvl);
    const float* regs = (lvl == 0 ? reg0 : (lvl == 1 ? reg1 : reg2)) + (size_t)idx * 4;
    float r0 = regs[0], r1 = regs[1], r2 = regs[2], r3 = regs[3];
    float sc = scales[lvl];
    int gx = (int)(idx % (u32)W), gy = (int)(idx / (u32)W);
    float cx = ((float)gx + 0.5f) * st + sigmoidf(r0) * 3.f - 1.5f;
    float cy = ((float)gy + 0.5f) * st + sigmoidf(r1) * 3.f - 1.5f;
    float bw = expf(r2 * sc), bh = expf(r3 * sc);
    float x1 = (cx - 0.5f * bw) * st, y1 = (cy - 0.5f * bh) * st;
    float x2 = (cx + 0.5f * bw) * st, y2 = (cy + 0.5f * bh) * st;
    int lab = labels_all[lvl_off(lvl) + (int)idx];

    float o = (float)lab * CLS_OFF;     // per-class separation (ref trick)
    obox[s * 4 + 0] = x1 + o; obox[s * 4 + 1] = y1 + o;
    obox[s * 4 + 2] = x2 + o; obox[s * 4 + 3] = y2 + o;
    valid[s] = (score >= CONF_T) ? 1u : 0u;
    gkeys[s] = ((u64)sb << 32) | (u64)(0xFFFFFFFFu - (u32)s);

    const float inv = 1.f / 2048.f;
    out[s * 4 + 0] = fminf(fmaxf(x1 * inv, 0.f), 1.f);
    out[s * 4 + 1] = fminf(fmaxf(y1 * inv, 0.f), 1.f);
    out[s * 4 + 2] = fminf(fmaxf(x2 * inv, 0.f), 1.f);
    out[s * 4 + 3] = fminf(fmaxf(y2 * inv, 0.f), 1.f);
    out[12000 + s] = score;
    out[15000 + s] = (float)lab;
}

// ---------------------------------------------------------------------------
// 5) gather into globally-sorted order (+ inert padding rows)
// ---------------------------------------------------------------------------
__global__ __launch_bounds__(256)
void gather_k(const u64* __restrict__ gkeys, const float* __restrict__ obox,
              const u32* __restrict__ valid, u32* __restrict__ order,
              float* __restrict__ sbox, u32* __restrict__ svalid) {
    int r = blockIdx.x * 256 + threadIdx.x;
    if (r >= N_PAD) return;
    if (r < N_CAND) {
        u64 k = gkeys[r];
        u32 slot = 0xFFFFFFFFu - (u32)(k & 0xFFFFFFFFull);
        order[r] = slot;
        ((float4*)sbox)[r] = ((const float4*)obox)[slot];
        svalid[r] = valid[slot];
    } else {
        ((float4*)sbox)[r] = make_float4(-3e8f, -3e8f, -3e8f, -3e8f);
        svalid[r] = 0u; order[r] = 0u;
    }
}

// ---------------------------------------------------------------------------
// 6) suppression bitmask.  94 balanced blocks (8 waves each); block b owns
//    row-bands b and 187-b (exactly 189 tiles).  The whole 48KB sorted-box
//    table is DMA'd to LDS once per block by the Tensor Data Mover, then all
//    tiles are LDS-only.  Per 16x16 tile, one V_WMMA_F32_16X16X4_F32 forms
//    area_i + area_j (A row = [area_i,1,0,0]; B col = [1,area_j,0,0]^T; the
//    hi-lane K=2/3 slots carry the zero padding).  Accumulator layout: VGPR k,
//    lanes0-15 = row k, lanes16-31 = row k+8, col = lane&15 -> ballot_w32 of
//    (iou>0.6 && j>i) gives rows k / k+8 in its lo/hi halves.
// ---------------------------------------------------------------------------
__global__ __launch_bounds__(256)
void mask_k(const float* __restrict__ sbox, u32* __restrict__ mask) {
    __shared__ float4 lbox[N_PAD];          // 48128 B staged box table
    int tid = threadIdx.x;
#if __has_builtin(__builtin_amdgcn_tensor_load_to_lds) && \
    __has_builtin(__builtin_amdgcn_s_wait_tensorcnt)
    if (tid < 32) {                         // wave 0 issues one bulk TDM load
        u32 lds_base = (u32)(unsigned long long)(void*)&lbox[0];
        unsigned long long ga = (unsigned long long)sbox;
        // D# group0: count=1, lds_addr, 57-bit global_addr, type=2 (image)
        u32x4 g0;
        g0[0] = 1u;
        g0[1] = lds_base;
        g0[2] = (u32)ga;
        g0[3] = ((u32)(ga >> 32) & 0x01FFFFFFu) | (2u << 30);
        // D# group1: 1-D tile of 12032 f32 elements (data_size=4B),
        // tensor_dim0 = tile_dim0 = 12032, tensor_dim1 = tile_dim1 = 1,
        // workgroup_mask=0 (not in cluster), no pad/iterate/barrier.
        i32x8 g1;
        g1[0] = (int)(2u << 16);                          // data_size=4B
        g1[1] = (int)((12032u & 0xFFFFu) << 16);          // tensor_dim0 lo16
        g1[2] = (int)((12032u >> 16) | (1u << 16));       // dim0 hi | dim1 lo
        g1[3] = (int)(12032u << 16);                      // dim1 hi | tile_dim0
        g1[4] = 1;                                        // tile_dim1=1, dim2=0
        g1[5] = 12032;                                    // dim0_stride lo32
        g1[6] = 0;                                        // stride hi | dim1_stride
        g1[7] = 0;
        i32x4 gz;  gz[0] = gz[1] = gz[2] = gz[3] = 0;     // groups 2/3 unused
        i32x8 gz8; gz8[0] = gz8[1] = gz8[2] = gz8[3] = 0;
        gz8[4] = gz8[5] = gz8[6] = gz8[7] = 0;
        // amdgpu-toolchain (clang-23) 6-arg form:
        // (g0 u32x4, g1 i32x8, g2 i32x4, g3 i32x4, extra i32x8, cpol i32)
        __builtin_amdgcn_tensor_load_to_lds(g0, g1, gz, gz, gz8, 0);
        __builtin_amdgcn_s_wait_tensorcnt(0);
    }
#else
    for (int t = tid; t < N_PAD; t += 256) lbox[t] = ((const float4*)sbox)[t];
#endif
    __syncthreads();

    int lane = tid & 31;
    int wave = tid >> 5;
    bool lo  = lane < 16;
    int  l15 = lane & 15;
    int  b   = blockIdx.x;

    for (int pass = 0; pass < 2; ++pass) {
        int ti = pass ? (NTILE - 1 - b) : b;
        // A fragment: row areas (reference computes area on OFFSET boxes)
        float4 ar = lo ? lbox[ti * 16 + lane] : make_float4(0.f, 0.f, 0.f, 0.f);
        float areaA = (ar.z - ar.x) * (ar.w - ar.y);
        v2f A;
        A.x = lo ? areaA : 0.f;     // K=0 (hi lanes: K=2 -> 0)
        A.y = lo ? 1.f   : 0.f;     // K=1 (hi lanes: K=3 -> 0)

        for (int tj = ti + wave; tj < NTILE; tj += 8) {
            float4 cb = lbox[tj * 16 + l15];
            float careaa = (cb.z - cb.x) * (cb.w - cb.y);
            v2f B;
            B.x = lo ? 1.f    : 0.f;
            B.y = lo ? careaa : 0.f;

#if __has_builtin(__builtin_amdgcn_wmma_f32_16x16x4_f32)
            v8f c = {};
            v8f sum = __builtin_amdgcn_wmma_f32_16x16x4_f32(
                false, A, false, B, (short)0, c, false, false);
#else
            v8f sum;
#pragma unroll
            for (int k = 0; k < 8; ++k) {
                int m = k + (lo ? 0 : 8);
                float4 rb = lbox[ti * 16 + m];
                sum[k] = (rb.z - rb.x) * (rb.w - rb.y) + careaa;
            }
#endif
            int wj = tj >> 1;
            int sh = (tj & 1) * 16;
#pragma unroll
            for (int k = 0; k < 8; ++k) {
                int m = k + (lo ? 0 : 8);
                float4 rb = lbox[ti * 16 + m];
                float xx1 = fmaxf(rb.x, cb.x);
                float yy1 = fmaxf(rb.y, cb.y);
                float xx2 = fminf(rb.z, cb.z);
                float yy2 = fminf(rb.w, cb.w);
                float w = fmaxf(1e-10f, xx2 - xx1);
                float h = fmaxf(1e-10f, yy2 - yy1);
                float inter = w * h;
                float iou = inter / (sum[k] - inter + 1e-14f);
                int gi = ti * 16 + m, gj = tj * 16 + l15;
                bool pred = (iou > NMS_T) && (gj > gi);
                u32 bal = __builtin_amdgcn_ballot_w32(pred);
                if (lane == 0) {
                    atomicOr(&mask[(ti * 16 + k)     * WORDS + wj], (bal & 0xFFFFu) << sh);
                    atomicOr(&mask[(ti * 16 + k + 8) * WORDS + wj], (bal >> 16)     << sh);
                }
            }
        }
    }
}

// ---------------------------------------------------------------------------
// 7) exact greedy-NMS reduction.  Single-wave workgroup: barriers lower to
//    waitcnt + S_NOP (no barrier resource for 1-wave groups), so the 3000
//    sequential rounds stay cheap.  Each lane owns 3 of the 94 mask words.
// ---------------------------------------------------------------------------
__global__ __launch_bounds__(32)
void reduce_k(const u32* __restrict__ mask, const u32* __restrict__ svalid,
              const u32* __restrict__ order, float* __restrict__ out) {
    __shared__ u32 rem[WORDS];
    int tid = threadIdx.x;
    for (int w = tid; w < WORDS; w += 32) {
        u32 v = 0u;
        for (int b = 0; b < 32; ++b) {
            int j = w * 32 + b;
            if (j >= N_CAND || !svalid[j]) v |= 1u << b;
        }
        rem[w] = v;
    }
    __syncthreads();
    for (int i = 0; i < N_CAND; ++i) {
        bool alive = ((rem[i >> 5] >> (i & 31)) & 1u) == 0u;  // uniform
        __syncthreads();
        if (alive)
            for (int w = tid; w < WORDS; w += 32)
                rem[w] |= mask[i * WORDS + w];
        __syncthreads();
    }
    for (int r = tid; r < N_CAND; r += 32) {
        bool keep = ((rem[r >> 5] >> (r & 31)) & 1u) == 0u;
        out[18000 + order[r]] = keep ? 1.f : 0.f;
    }
}

// ---------------------------------------------------------------------------
extern "C" void kernel_launch(void* const* d_in, const int* in_sizes, int n_in,
                              void* d_out, int out_size, void* d_ws, size_t ws_size,
                              hipStream_t stream) {
    const float* cls0   = (const float*)d_in[0];
    const float* reg0   = (const float*)d_in[1];
    const float* cls1   = (const float*)d_in[2];
    const float* reg1   = (const float*)d_in[3];
    const float* cls2   = (const float*)d_in[4];
    const float* reg2   = (const float*)d_in[5];
    const float* scales = (const float*)d_in[6];
    float* out = (float*)d_out;

    char* ws = (char*)d_ws;
    float* scores  = (float*)(ws + OFF_SCORES);
    int*   labels  = (int*)  (ws + OFF_LABELS);
    u32*   hist    = (u32*)  (ws + OFF_HIST);
    u64*   prefix  = (u64*)  (ws + OFF_STATE);          // [3]
    u32*   krem    = (u32*)  (ws + OFF_STATE + 24);     // [3]
    u32*   cnt     = (u32*)  (ws + OFF_STATE + 36);     // [3]
    u64*   sel     = (u64*)  (ws + OFF_SEL);
    float* obox    = (float*)(ws + OFF_OBOX);
    u32*   valid   = (u32*)  (ws + OFF_VALID);
    u64*   gkeys   = (u64*)  (ws + OFF_GKEYS);
    u32*   order   = (u32*)  (ws + OFF_ORDER);
    float* sbox    = (float*)(ws + OFF_SBOX);
    u32*   svalid  = (u32*)  (ws + OFF_SVALID);
    u32*   mask    = (u32*)  (ws + OFF_MASK);

    init_ws<<<512, 256, 0, stream>>>(hist, cnt, prefix, krem, sel, gkeys, mask);
    score_kernel<<<(N_ANCH + 255) / 256, 256, 0, stream>>>(cls0, cls1, cls2, scores, labels);
    for (int p = 0; p < 8; ++p) {
        radix_hist<<<dim3(256, 3), 256, 0, stream>>>(scores, prefix, hist, p);
        radix_scan<<<1, 32, 0, stream>>>(hist, prefix, krem, p);
    }
    compact_k<<<dim3(256, 3), 256, 0, stream>>>(scores, prefix, cnt, sel);
    sort1024<<<3, 512, 0, stream>>>(sel);
    decode_k<<<(N_CAND + 255) / 256, 256, 0, stream>>>(sel, reg0, reg1, reg2, scales,
                                                       labels, obox, valid, gkeys, out);
    gsort<<<1, 1024, 0, stream>>>(gkeys);
    gather_k<<<(N_PAD + 255) / 256, 256, 0, stream>>>(gkeys, obox, valid,
                                                      order, sbox, svalid);
    mask_k<<<NTILE / 2, 256, 0, stream>>>(sbox, mask);
    reduce_k<<<1, 32, 0, stream>>>(mask, svalid, order, out);
}